// BiquadFilter_73160472920667
// MI455X (gfx1250) — compile-verified
//
#include <hip/hip_runtime.h>
#include <hip/hip_bf16.h>

// ---------------------------------------------------------------------------
// Biquad LPF (Direct Form 2 Transposed) as a block-parallel linear recurrence.
//
//   Pass 1 (WMMA): zero-state block responses  Y16x16 = H(16x16) * X(16x16)
//                  via 4x V_WMMA_F32_16X16X4_F32, + per-block exit states.
//                  Tile staging uses GLOBAL_LOAD_ASYNC_TO_LDS_B128 (ASYNCcnt)
//                  when the toolchain exposes it.
//   Pass 2 (scan): compose per-block 2x2 affine state maps along each row.
//   Pass 3 (fixup): y[t] += f1[t%16]*w1_in + f2[t%16]*w2_in   (bandwidth pass)
// ---------------------------------------------------------------------------

typedef __attribute__((ext_vector_type(2))) float v2f;
typedef __attribute__((ext_vector_type(8))) float v8f;
typedef __attribute__((ext_vector_type(4))) int   v4i;

#if defined(__has_builtin)
#if __has_builtin(__builtin_amdgcn_global_load_async_to_lds_b128)
#define HAVE_ASYNC_TO_LDS 1
#endif
#endif

#define B_ROWS   32
#define T_LEN    262144
#define BLK      16                 // samples per block (WMMA tile edge)
#define NBLK     (T_LEN / BLK)      // 16384 blocks per row
#define TILE     256                // samples per wave (16 blocks)
#define TILES_PER_ROW (T_LEN / TILE)// 1024
#define WAVES_PER_WG  8
#define SCAN_LANES    32
#define CHUNK    (NBLK / SCAN_LANES) // 512 blocks per scan lane

// c = {b0, b1, b2, a1, a2} (normalized by a0), Audio EQ Cookbook LPF.
__device__ __forceinline__ void lpf_coeffs(int freq_i, float* c) {
    double f = (double)freq_i;
    if (f < 20.0)    f = 20.0;
    if (f > 21950.0) f = 21950.0;              // 44100/2 - 100
    const double Q  = 0.707;                   // clip(0.707, 0.1, 30)
    double w0 = 2.0 * 3.14159265358979323846 * f / 44100.0;
    double sw = sin(w0), cw = cos(w0);
    double alpha = sw / (2.0 * Q);
    double a0 = 1.0 + alpha;
    c[0] = (float)(((1.0 - cw) * 0.5) / a0);
    c[1] = (float)((1.0 - cw) / a0);
    c[2] = (float)(((1.0 - cw) * 0.5) / a0);
    c[3] = (float)((-2.0 * cw) / a0);
    c[4] = (float)((1.0 - alpha) / a0);
}

// One DF2T step. y = b0*x + w1; w1' = b1*x - a1*y + w2; w2' = b2*x - a2*y
__device__ __forceinline__ float df2t_step(const float* c, float x,
                                           float& w1, float& w2) {
    float y   = c[0] * x + w1;
    float w1n = c[1] * x - c[3] * y + w2;
    float w2n = c[2] * x - c[4] * y;
    w1 = w1n; w2 = w2n;
    return y;
}

// ---------------------------------------------------------------------------
// Kernel 1: zero-state block outputs via WMMA + per-block exit states.
// Grid: (TILES_PER_ROW/8, B_ROWS) x 256 threads (8 waves); 1 tile per wave.
// ---------------------------------------------------------------------------
__global__ __launch_bounds__(256)
void biquad_k1_wmma(const float* __restrict__ x, const int* __restrict__ freqp,
                    float* __restrict__ y, float2* __restrict__ s0) {
    __shared__ float sH[16 * 20];            // padded lower-triangular H
    __shared__ float sxs[WAVES_PER_WG * 320];// padded x tiles (stride 20/row)
    __shared__ float sh[16];                 // impulse response
    __shared__ float sc[5];                  // coefficients

    const int tid = threadIdx.x;
    if (tid == 0) {
        float c[5];
        lpf_coeffs(*freqp, c);
        #pragma unroll
        for (int i = 0; i < 5; i++) sc[i] = c[i];
        float w1 = 0.f, w2 = 0.f;
        for (int n = 0; n < 16; n++)
            sh[n] = df2t_step(c, (n == 0) ? 1.f : 0.f, w1, w2);
    }
    __syncthreads();
    { // fill H[i][j] = h[i-j] (i>=j), one entry per thread
        int i = tid >> 4, j = tid & 15;
        sH[i * 20 + j] = (i >= j) ? sh[i - j] : 0.f;
    }

    const int wave = tid >> 5;
    const int lane = tid & 31;
    const int row  = blockIdx.y;
    const int tile = blockIdx.x * WAVES_PER_WG + wave;

    const float* xt = x + (size_t)row * T_LEN + (size_t)tile * TILE;
    float* xs = &sxs[wave * 320];

    // stage 256 contiguous samples -> padded LDS (element i at (i>>4)*20+(i&15))
    {
        int p    = lane * 8;
        int lidx = (p >> 4) * 20 + (p & 15);      // 16B-aligned (20 % 4 == 0)
#ifdef HAVE_ASYNC_TO_LDS
        typedef __attribute__((address_space(1))) v4i gv4i;
        typedef __attribute__((address_space(3))) v4i lv4i;
        gv4i* gp = (gv4i*)(xt + p);
        lv4i* lp = (lv4i*)(xs + lidx);
        // IOFFSET applies to both global and LDS addresses (ISA 10, async ops)
        __builtin_amdgcn_global_load_async_to_lds_b128(gp, lp, 0,  0);
        __builtin_amdgcn_global_load_async_to_lds_b128(gp, lp, 16, 0);
#if __has_builtin(__builtin_amdgcn_s_wait_asynccnt)
        __builtin_amdgcn_s_wait_asynccnt(0);
#else
        asm volatile("s_wait_asynccnt 0x0" ::: "memory");
#endif
#else
        float4 v0 = *(const float4*)(xt + p);
        float4 v1 = *(const float4*)(xt + p + 4);
        *(float4*)(xs + lidx)     = v0;
        *(float4*)(xs + lidx + 4) = v1;
#endif
    }
    __syncthreads();   // orders H fill + x staging for all waves

    const int c16 = lane & 15;   // A row m / B,C column n
    const int hi  = lane >> 4;   // K-half (A/B) and M-half (C/D)

    // D = H * X accumulated over four K=4 slices.
    v8f acc = {0.f, 0.f, 0.f, 0.f, 0.f, 0.f, 0.f, 0.f};
    #pragma unroll
    for (int kb = 0; kb < 16; kb += 4) {
        v2f A, Bv;
        A[0]  = sH[c16 * 20 + kb + 2 * hi];        // A[m][kb + (hi?2:0)]
        A[1]  = sH[c16 * 20 + kb + 1 + 2 * hi];    // A[m][kb + (hi?3:1)]
        Bv[0] = xs[c16 * 20 + kb + 2 * hi];        // X[kb + (hi?2:0)][n]
        Bv[1] = xs[c16 * 20 + kb + 1 + 2 * hi];    // X[kb + (hi?3:1)][n]
        acc = __builtin_amdgcn_wmma_f32_16x16x4_f32(
            /*neg_a=*/false, A, /*neg_b=*/false, Bv,
            /*c_mod=*/(short)0, acc, /*reuse_a=*/false, /*reuse_b=*/false);
    }

    // Lane holds block c16, samples hi*8 .. hi*8+7 -> contiguous B128 stores.
    float* yo = y + (size_t)row * T_LEN + (size_t)tile * TILE + c16 * 16 + hi * 8;
    float4 o0 = {acc[0], acc[1], acc[2], acc[3]};
    float4 o1 = {acc[4], acc[5], acc[6], acc[7]};
    *(float4*)yo       = o0;
    *((float4*)yo + 1) = o1;

    // Zero-state exit state from x14,x15,y14,y15 (lanes 16..31 hold y14,y15).
    if (hi) {
        float y14 = acc[6], y15 = acc[7];
        float x14 = xs[c16 * 20 + 14];
        float x15 = xs[c16 * 20 + 15];
        float w2_after14 = sc[2] * x14 - sc[4] * y14;
        float w2o = sc[2] * x15 - sc[4] * y15;
        float w1o = sc[1] * x15 - sc[3] * y15 + w2_after14;
        s0[(size_t)row * NBLK + (size_t)tile * 16 + c16] = make_float2(w1o, w2o);
    }
}

// ---------------------------------------------------------------------------
// Kernel 2: scan of per-block affine state maps: s[k+1] = A16*s[k] + s0[k].
// One wave per row; each lane serially composes 512 maps, wave prefix, then
// rewrites ws in place with the *entry* state of every block.
// ---------------------------------------------------------------------------
__global__ __launch_bounds__(32)
void biquad_k2_scan(const int* __restrict__ freqp, float2* __restrict__ s) {
    __shared__ float  pm[SCAN_LANES * 6];
    __shared__ float2 pref[SCAN_LANES];

    const int row  = blockIdx.x;
    const int lane = threadIdx.x;

    float c[5];
    lpf_coeffs(*freqp, c);

    // A16 = (zero-input step matrix [[-a1,1],[-a2,0]])^16, via two columns.
    float e00 = 1.f, e10 = 0.f, e01 = 0.f, e11 = 1.f;
    #pragma unroll
    for (int n = 0; n < 16; n++) {
        float n00 = -c[3] * e00 + e10, n10 = -c[4] * e00;
        float n01 = -c[3] * e01 + e11, n11 = -c[4] * e01;
        e00 = n00; e10 = n10; e01 = n01; e11 = n11;
    }
    const float a00 = e00, a01 = e01, a10 = e10, a11 = e11;

    float2* srow = s + (size_t)row * NBLK;
    const int base = lane * CHUNK;

    // Pass 1: compose this lane's chunk: M_acc, v_acc with s -> M s + v.
    float M00 = 1.f, M01 = 0.f, M10 = 0.f, M11 = 1.f, vx = 0.f, vy = 0.f;
    for (int b = 0; b < CHUNK; b++) {
        float2 q = srow[base + b];
        float n00 = a00 * M00 + a01 * M10, n01 = a00 * M01 + a01 * M11;
        float n10 = a10 * M00 + a11 * M10, n11 = a10 * M01 + a11 * M11;
        float nvx = a00 * vx + a01 * vy + q.x;
        float nvy = a10 * vx + a11 * vy + q.y;
        M00 = n00; M01 = n01; M10 = n10; M11 = n11; vx = nvx; vy = nvy;
    }
    pm[lane * 6 + 0] = M00; pm[lane * 6 + 1] = M01;
    pm[lane * 6 + 2] = M10; pm[lane * 6 + 3] = M11;
    pm[lane * 6 + 4] = vx;  pm[lane * 6 + 5] = vy;
    __syncthreads();

    if (lane == 0) {       // serial 32-step prefix (entry state of each chunk)
        float px = 0.f, py = 0.f;
        for (int l = 0; l < SCAN_LANES; l++) {
            pref[l] = make_float2(px, py);
            float nx = pm[l * 6 + 0] * px + pm[l * 6 + 1] * py + pm[l * 6 + 4];
            float ny = pm[l * 6 + 2] * px + pm[l * 6 + 3] * py + pm[l * 6 + 5];
            px = nx; py = ny;
        }
    }
    __syncthreads();

    // Pass 2: rewrite ws in place with each block's entry state.
    float sx = pref[lane].x, sy = pref[lane].y;
    for (int b = 0; b < CHUNK; b++) {
        float2 q = srow[base + b];
        srow[base + b] = make_float2(sx, sy);
        float nx = a00 * sx + a01 * sy + q.x;
        float ny = a10 * sx + a11 * sy + q.y;
        sx = nx; sy = ny;
    }
}

// ---------------------------------------------------------------------------
// Kernel 3: y[t] += f1[t%16]*w1_in + f2[t%16]*w2_in. Pure bandwidth pass.
// ---------------------------------------------------------------------------
__global__ __launch_bounds__(256)
void biquad_k3_fixup(const int* __restrict__ freqp,
                     const float2* __restrict__ s, float* __restrict__ y) {
    __shared__ float f1[16], f2[16];
    if (threadIdx.x == 0) {
        float c[5];
        lpf_coeffs(*freqp, c);
        float w1 = 1.f, w2 = 0.f;
        for (int n = 0; n < 16; n++) f1[n] = df2t_step(c, 0.f, w1, w2);
        w1 = 0.f; w2 = 1.f;
        for (int n = 0; n < 16; n++) f2[n] = df2t_step(c, 0.f, w1, w2);
    }
    __syncthreads();

    size_t t4 = ((size_t)blockIdx.x * blockDim.x + threadIdx.x) * 4;
    float2 st = s[t4 >> 4];          // (row*T + tt)/16 == row*NBLK + blk
    int i0 = (int)(t4 & 15);         // 0,4,8,12

    float4 yv = *(float4*)(y + t4);
    yv.x += f1[i0 + 0] * st.x + f2[i0 + 0] * st.y;
    yv.y += f1[i0 + 1] * st.x + f2[i0 + 1] * st.y;
    yv.z += f1[i0 + 2] * st.x + f2[i0 + 2] * st.y;
    yv.w += f1[i0 + 3] * st.x + f2[i0 + 3] * st.y;
    *(float4*)(y + t4) = yv;
}

// ---------------------------------------------------------------------------
extern "C" void kernel_launch(void* const* d_in, const int* in_sizes, int n_in,
                              void* d_out, int out_size, void* d_ws, size_t ws_size,
                              hipStream_t stream) {
    const float* x    = (const float*)d_in[0];
    const int*   freq = (const int*)d_in[1];
    float*       y    = (float*)d_out;
    float2*      s    = (float2*)d_ws;   // 32 * 16384 * 8 B = 4 MB

    dim3 g1(TILES_PER_ROW / WAVES_PER_WG, B_ROWS);   // (128, 32) x 256
    biquad_k1_wmma<<<g1, 256, 0, stream>>>(x, freq, y, s);

    biquad_k2_scan<<<B_ROWS, SCAN_LANES, 0, stream>>>(freq, s);

    int n4 = (B_ROWS * T_LEN) / 4;
    biquad_k3_fixup<<<n4 / 256, 256, 0, stream>>>(freq, s, y);
}